// ABL_SPARSE_87694642250045
// MI455X (gfx1250) — compile-verified
//
#include <hip/hip_runtime.h>
#include <math.h>

// ---------------------------------------------------------------------------
// Graph matching network on MI455X (gfx1250).
// ~40 GFLOP total, ~50MB working set (L2-resident): latency-bound, so keep
// fp32 and use V_WMMA_F32_16X16X4_F32 for every GEMM. Gathers, rank-1 edge
// term, bias, relu, residual, and segment_sum (atomic scatter) are all fused
// into the WMMA GEMM epilogue. Sinkhorns run per-graph-pair in LDS.
// GEMM kernel is templated on N so W-operand loads use immediate offsets,
// and uses two independent WMMA accumulator chains for matrix-pipe ILP.
// ---------------------------------------------------------------------------

#define B_       64
#define N_NODES  32
#define E_EDGES  96
#define G_       (2 * B_)            // 128 graphs
#define NT_      (G_ * N_NODES)      // 4096 nodes
#define NE_      (G_ * E_EDGES)      // 12288 edges
#define H_DIM    128
#define MSG_DIM  256
#define TEMP_    0.1f
#define SK_ITERS 20

typedef __attribute__((ext_vector_type(2))) float v2f;
typedef __attribute__((ext_vector_type(8))) float v8f;

struct GemmDesc {
  const float* A0;      // first K segment, rows gathered via g0 (or identity)
  const float* A1;      // second K segment (nullptr if unused)
  int lda0, lda1;
  int K0, K;            // K0 = split point, K = total dense K (segments mult of 8)
  const int* g0;        // row gather for A0 (nullptr = identity)
  const int* g1;        // row gather for A1
  const float* W;       // K x N row-major
  const float* bias;    // length N (nullable)
  const float* r1e;     // per-row scalar (edge feature), nullable
  const float* r1w;     // length-N weight row for rank-1 term, nullable
  const float* resid;   // residual added post-activation, nullable
  int ldresid;
  float* C;
  int ldc;
  const int* scatter;   // if set: atomicAdd into row scatter[m] (segment_sum)
  int M, N;             // multiples of 16
  int relu;
  int accum;            // C += val instead of C = val
};

__device__ inline v8f wmma_f32(v2f a, v2f b, v8f c) {
  return __builtin_amdgcn_wmma_f32_16x16x4_f32(false, a, false, b,
                                               (short)0, c, false, false);
}

// Accumulate one concat segment: `len` K-steps (multiple of 8).
// ap: this lane's A row at K-offset (2*half); strides +4 per WMMA.
// wp: W column (tn+lr) at K-offset (2*half); strides +4*N_ per WMMA.
// Two accumulator chains (acc0/acc1) break the WMMA RAW dependency chain.
template <int N_>
__device__ inline void gemm_segment(v8f& acc0, v8f& acc1,
                                    const float* ap, const float* wp, int len) {
  for (int k = 0; k < len; k += 8) {
    const float2 a0  = *(const float2*)ap;
    const float  b0x = wp[0];
    const float  b0y = wp[N_];
    const float2 a1  = *(const float2*)(ap + 4);
    const float  b1x = wp[4 * N_];
    const float  b1y = wp[5 * N_];
    v2f a, b;
    a.x = a0.x; a.y = a0.y; b.x = b0x; b.y = b0y;
    acc0 = wmma_f32(a, b, acc0);
    a.x = a1.x; a.y = a1.y; b.x = b1x; b.y = b1y;
    acc1 = wmma_f32(a, b, acc1);
    ap += 8;
    wp += 8 * N_;
  }
}

// One wave computes one 16x16 fp32 tile with V_WMMA_F32_16X16X4_F32.
// A 16x4 layout: lanes 0-15 -> M, VGPR0/1 hold K=(2*half),(2*half+1).
// B 4x16 layout mirrors A with N across lanes.
// C/D: acc[v] holds (M = tm + v + 8*half, N = tn + (lane&15)).
template <int N_>
__global__ __launch_bounds__(256) void wmma_gemm_kernel(GemmDesc d) {
  const int wave   = blockIdx.x * (blockDim.x >> 5) + (threadIdx.x >> 5);
  const int tilesN = N_ >> 4;
  const int tiles  = (d.M >> 4) * tilesN;
  if (wave >= tiles) return;                    // uniform per wave: EXEC all-1s
  const int tm   = (wave / tilesN) << 4;
  const int tn   = (wave % tilesN) << 4;
  const int lane = threadIdx.x & 31;
  const int half = lane >> 4;
  const int lr   = lane & 15;
  const int arow = tm + lr;
  const int kh   = half << 1;                   // this lane-half's K offset

  v8f acc0 = {};
  v8f acc1 = {};
  {
    const int r0 = d.g0 ? d.g0[arow] : arow;
    const float* ap = d.A0 + (size_t)r0 * d.lda0 + kh;
    const float* wp = d.W + (size_t)kh * N_ + tn + lr;
    gemm_segment<N_>(acc0, acc1, ap, wp, d.K0);
  }
  if (d.A1) {
    const int r1 = d.g1 ? d.g1[arow] : arow;
    const float* ap = d.A1 + (size_t)r1 * d.lda1 + kh;
    const float* wp = d.W + (size_t)(d.K0 + kh) * N_ + tn + lr;
    gemm_segment<N_>(acc0, acc1, ap, wp, d.K - d.K0);
  }
  const v8f acc = acc0 + acc1;

  const int   n    = tn + lr;
  const float bv   = d.bias ? d.bias[n] : 0.0f;
  const float r1wn = d.r1w ? d.r1w[n] : 0.0f;
#pragma unroll
  for (int v = 0; v < 8; ++v) {
    const int m = tm + v + (half << 3);
    float val = acc[v] + bv;
    if (d.r1e)  val += d.r1e[m] * r1wn;
    if (d.relu) val = fmaxf(val, 0.0f);
    if (d.resid) val += d.resid[(size_t)m * d.ldresid + n];
    if (d.scatter) {
      atomicAdd(&d.C[(size_t)d.scatter[m] * d.ldc + n], val);
    } else if (d.accum) {
      d.C[(size_t)m * d.ldc + n] += val;
    } else {
      d.C[(size_t)m * d.ldc + n] = val;
    }
  }
}

__global__ void zero_kernel(float* p, int n) {
  int i = blockIdx.x * blockDim.x + threadIdx.x;
  if (i < n) p[i] = 0.0f;
}

// Per graph-pair: cost(L1 of 32-dim sinkhorn features) -> 20-iter sinkhorn in
// LDS -> node_plan (saved) and node alignment (plan-weighted L1 over 128-dim h).
__global__ __launch_bounds__(256) void node_plan_kernel(
    const float* __restrict__ t, const float* __restrict__ h,
    float* __restrict__ plan, float* __restrict__ out) {
  const int g   = blockIdx.x;
  const int tid = threadIdx.x;
  __shared__ float tq[32 * 32];
  __shared__ float tc[32 * 32];
  __shared__ float la[32 * 32];
  __shared__ float red[256];
  const int qbase = (2 * g) * 32;
  const int cbase = (2 * g + 1) * 32;

  for (int idx = tid; idx < 1024; idx += 256) {
    tq[idx] = t[qbase * 32 + idx];
    tc[idx] = t[cbase * 32 + idx];
  }
  __syncthreads();
  for (int idx = tid; idx < 1024; idx += 256) {
    const int i = idx >> 5, j = idx & 31;
    float s = 0.0f;
    for (int d = 0; d < 32; ++d) s += fabsf(tq[i * 32 + d] - tc[j * 32 + d]);
    la[idx] = -s * (1.0f / TEMP_);
  }
  __syncthreads();
  for (int it = 0; it < SK_ITERS; ++it) {
    if (tid < 32) {                      // normalize over j (axis=2)
      const int i = tid;
      float m = -3.4e38f;
      for (int j = 0; j < 32; ++j) m = fmaxf(m, la[i * 32 + j]);
      float s = 0.0f;
      for (int j = 0; j < 32; ++j) s += expf(la[i * 32 + j] - m);
      const float l = m + logf(s);
      for (int j = 0; j < 32; ++j) la[i * 32 + j] -= l;
    }
    __syncthreads();
    if (tid < 32) {                      // normalize over i (axis=1)
      const int j = tid;
      float m = -3.4e38f;
      for (int i = 0; i < 32; ++i) m = fmaxf(m, la[i * 32 + j]);
      float s = 0.0f;
      for (int i = 0; i < 32; ++i) s += expf(la[i * 32 + j] - m);
      const float l = m + logf(s);
      for (int i = 0; i < 32; ++i) la[i * 32 + j] -= l;
    }
    __syncthreads();
  }
  for (int idx = tid; idx < 1024; idx += 256) {
    const float pv = expf(la[idx]);
    la[idx] = pv;
    plan[g * 1024 + idx] = pv;
  }
  __syncthreads();
  float accv = 0.0f;
  for (int idx = tid; idx < 1024; idx += 256) {
    const int i = idx >> 5, j = idx & 31;
    const float* hq = h + (size_t)(qbase + i) * H_DIM;
    const float* hc = h + (size_t)(cbase + j) * H_DIM;
    float s = 0.0f;
    for (int d = 0; d < H_DIM; ++d) s += fabsf(hq[d] - hc[d]);
    accv += la[idx] * s;
  }
  red[tid] = accv;
  __syncthreads();
  for (int off = 128; off > 0; off >>= 1) {
    if (tid < off) red[tid] += red[tid + off];
    __syncthreads();
  }
  if (tid == 0) out[g] = red[0];
}

// Per graph-pair: kron of node_plan -> 96x96 sinkhorn in LDS -> edge alignment
// (plan-weighted L1 over 256-dim edge embeddings). Accumulates into out[g].
__global__ __launch_bounds__(256) void edge_plan_kernel(
    const float* __restrict__ plan, const float* __restrict__ em,
    const int* __restrict__ from_idx, const int* __restrict__ to_idx,
    float* __restrict__ out) {
  const int g   = blockIdx.x;
  const int tid = threadIdx.x;
  __shared__ float P[32 * 32];
  __shared__ float la[96 * 96];
  __shared__ int qf[96], qt[96], cf[96], ct[96];
  __shared__ float red[256];
  const int qg = 2 * g, cg = 2 * g + 1;

  for (int idx = tid; idx < 1024; idx += 256) P[idx] = plan[g * 1024 + idx];
  for (int e = tid; e < 96; e += 256) {
    qf[e] = from_idx[qg * 96 + e] - qg * 32;
    qt[e] = to_idx[qg * 96 + e] - qg * 32;
    cf[e] = from_idx[cg * 96 + e] - cg * 32;
    ct[e] = to_idx[cg * 96 + e] - cg * 32;
  }
  __syncthreads();
  for (int idx = tid; idx < 96 * 96; idx += 256) {
    const int i = idx / 96, j = idx % 96;
    const float straight = P[qf[i] * 32 + cf[j]] * P[qt[i] * 32 + ct[j]];
    const float cross    = P[qf[i] * 32 + ct[j]] * P[qt[i] * 32 + cf[j]];
    la[idx] = fmaxf(straight, cross) * (1.0f / TEMP_);
  }
  __syncthreads();
  for (int it = 0; it < SK_ITERS; ++it) {
    if (tid < 96) {
      const int i = tid;
      float m = -3.4e38f;
      for (int j = 0; j < 96; ++j) m = fmaxf(m, la[i * 96 + j]);
      float s = 0.0f;
      for (int j = 0; j < 96; ++j) s += expf(la[i * 96 + j] - m);
      const float l = m + logf(s);
      for (int j = 0; j < 96; ++j) la[i * 96 + j] -= l;
    }
    __syncthreads();
    if (tid < 96) {
      const int j = tid;
      float m = -3.4e38f;
      for (int i = 0; i < 96; ++i) m = fmaxf(m, la[i * 96 + j]);
      float s = 0.0f;
      for (int i = 0; i < 96; ++i) s += expf(la[i * 96 + j] - m);
      const float l = m + logf(s);
      for (int i = 0; i < 96; ++i) la[i * 96 + j] -= l;
    }
    __syncthreads();
  }
  float accv = 0.0f;
  for (int idx = tid; idx < 96 * 96; idx += 256) {
    const int i = idx / 96, j = idx % 96;
    const float pv = expf(la[idx]);
    const float* qe = em + (size_t)(qg * 96 + i) * MSG_DIM;
    const float* ce = em + (size_t)(cg * 96 + j) * MSG_DIM;
    float s = 0.0f;
    for (int d = 0; d < MSG_DIM; ++d) s += fabsf(qe[d] - ce[d]);
    accv += pv * s;
  }
  red[tid] = accv;
  __syncthreads();
  for (int off = 128; off > 0; off >>= 1) {
    if (tid < off) red[tid] += red[tid + off];
    __syncthreads();
  }
  if (tid == 0) out[g] += red[0];   // node_plan_kernel wrote out[g] first
}

static void launch_gemm(const GemmDesc& d, hipStream_t s) {
  const int tiles  = (d.M / 16) * (d.N / 16);
  const int blocks = (tiles + 7) / 8;            // 8 waves / 256-thread block
  switch (d.N) {
    case 32:  wmma_gemm_kernel<32><<<blocks, 256, 0, s>>>(d);  break;
    case 128: wmma_gemm_kernel<128><<<blocks, 256, 0, s>>>(d); break;
    case 256: wmma_gemm_kernel<256><<<blocks, 256, 0, s>>>(d); break;
    default:  break;  // unreachable for this network
  }
}

extern "C" void kernel_launch(void* const* d_in, const int* in_sizes, int n_in,
                              void* d_out, int out_size, void* d_ws, size_t ws_size,
                              hipStream_t stream) {
  const float* node_features = (const float*)d_in[0];
  const float* edge_features = (const float*)d_in[1];
  const float* W_enc  = (const float*)d_in[2];
  const float* b_enc  = (const float*)d_in[3];
  const float* W_msg1 = (const float*)d_in[4];
  const float* b_msg1 = (const float*)d_in[5];
  const float* W_msg2 = (const float*)d_in[6];
  const float* b_msg2 = (const float*)d_in[7];
  const float* W_upd  = (const float*)d_in[8];
  const float* b_upd  = (const float*)d_in[9];
  const float* W_sk1  = (const float*)d_in[10];
  const float* b_sk1  = (const float*)d_in[11];
  const float* W_sk2  = (const float*)d_in[12];
  const float* b_sk2  = (const float*)d_in[13];
  const float* W_lrl1 = (const float*)d_in[14];
  const float* b_lrl1 = (const float*)d_in[15];
  const float* W_lrl2 = (const float*)d_in[16];
  const float* b_lrl2 = (const float*)d_in[17];
  const int* from_idx = (const int*)d_in[18];
  const int* to_idx   = (const int*)d_in[19];
  float* out = (float*)d_out;

  float* p = (float*)d_ws;
  float* h0     = p; p += (size_t)NT_ * H_DIM;
  float* h1     = p; p += (size_t)NT_ * H_DIM;
  float* agg    = p; p += (size_t)NT_ * MSG_DIM;
  float* hidden = p; p += (size_t)NE_ * MSG_DIM;
  float* em     = p; p += (size_t)NE_ * MSG_DIM;
  float* th     = p; p += (size_t)NT_ * 32;
  float* tt     = p; p += (size_t)NT_ * 32;
  float* plan   = p; p += (size_t)B_ * 32 * 32;

  // ---- encoder: h = node_features @ W_enc + b_enc ----
  {
    GemmDesc d{};
    d.A0 = node_features; d.lda0 = 32; d.K0 = 32; d.K = 32;
    d.W = W_enc; d.bias = b_enc; d.N = H_DIM;
    d.C = h0; d.ldc = H_DIM; d.M = NT_;
    launch_gemm(d, stream);
  }

  float* cur = h0;
  float* nxt = h1;
  for (int prop = 0; prop < 5; ++prop) {
    zero_kernel<<<(NT_ * MSG_DIM + 255) / 256, 256, 0, stream>>>(agg, NT_ * MSG_DIM);
    for (int dir = 0; dir < 2; ++dir) {
      const int* gA = dir == 0 ? from_idx : to_idx;   // first 128 dims
      const int* gB = dir == 0 ? to_idx : from_idx;   // next 128 dims
      const int* sc = dir == 0 ? to_idx : from_idx;   // segment_sum target
      // layer 1: hidden = relu(concat[h[gA], h[gB]] @ W1[:256] + e*W1[256] + b1)
      GemmDesc d1{};
      d1.A0 = cur; d1.lda0 = H_DIM; d1.g0 = gA;
      d1.A1 = cur; d1.lda1 = H_DIM; d1.g1 = gB;
      d1.K0 = H_DIM; d1.K = 2 * H_DIM;
      d1.W = W_msg1; d1.bias = b_msg1; d1.N = MSG_DIM;
      d1.r1e = edge_features; d1.r1w = W_msg1 + 256 * MSG_DIM;
      d1.C = hidden; d1.ldc = MSG_DIM; d1.M = NE_; d1.relu = 1;
      launch_gemm(d1, stream);
      // layer 2 fused with segment_sum: agg[sc[m]] += hidden @ W2 + b2
      GemmDesc d2{};
      d2.A0 = hidden; d2.lda0 = MSG_DIM; d2.K0 = MSG_DIM; d2.K = MSG_DIM;
      d2.W = W_msg2; d2.bias = b_msg2; d2.N = MSG_DIM;
      d2.C = agg; d2.ldc = MSG_DIM; d2.M = NE_; d2.scatter = sc;
      launch_gemm(d2, stream);
    }
    // update: h' = h + concat[h, agg] @ W_upd + b_upd
    GemmDesc du{};
    du.A0 = cur; du.lda0 = H_DIM;
    du.A1 = agg; du.lda1 = MSG_DIM;
    du.K0 = H_DIM; du.K = H_DIM + MSG_DIM;
    du.W = W_upd; du.bias = b_upd; du.N = H_DIM;
    du.resid = cur; du.ldresid = H_DIM;
    du.C = nxt; du.ldc = H_DIM; du.M = NT_;
    launch_gemm(du, stream);
    float* tmp = cur; cur = nxt; nxt = tmp;
  }

  // ---- sinkhorn feature MLP: t = relu(h@Wsk1+b) @ Wsk2 + b ----
  {
    GemmDesc d{};
    d.A0 = cur; d.lda0 = H_DIM; d.K0 = H_DIM; d.K = H_DIM;
    d.W = W_sk1; d.bias = b_sk1; d.N = 32; d.relu = 1;
    d.C = th; d.ldc = 32; d.M = NT_;
    launch_gemm(d, stream);
    GemmDesc d2{};
    d2.A0 = th; d2.lda0 = 32; d2.K0 = 32; d2.K = 32;
    d2.W = W_sk2; d2.bias = b_sk2; d2.N = 32;
    d2.C = tt; d2.ldc = 32; d2.M = NT_;
    launch_gemm(d2, stream);
  }
  node_plan_kernel<<<B_, 256, 0, stream>>>(tt, cur, plan, out);

  // ---- edge embeddings: em = mlp2(f,t) + mlp2(t,f) with W_lrl ----
  for (int dir = 0; dir < 2; ++dir) {
    const int* gA = dir == 0 ? from_idx : to_idx;
    const int* gB = dir == 0 ? to_idx : from_idx;
    GemmDesc d1{};
    d1.A0 = cur; d1.lda0 = H_DIM; d1.g0 = gA;
    d1.A1 = cur; d1.lda1 = H_DIM; d1.g1 = gB;
    d1.K0 = H_DIM; d1.K = 2 * H_DIM;
    d1.W = W_lrl1; d1.bias = b_lrl1; d1.N = MSG_DIM;
    d1.r1e = edge_features; d1.r1w = W_lrl1 + 256 * MSG_DIM;
    d1.C = hidden; d1.ldc = MSG_DIM; d1.M = NE_; d1.relu = 1;
    launch_gemm(d1, stream);
    GemmDesc d2{};
    d2.A0 = hidden; d2.lda0 = MSG_DIM; d2.K0 = MSG_DIM; d2.K = MSG_DIM;
    d2.W = W_lrl2; d2.bias = b_lrl2; d2.N = MSG_DIM;
    d2.C = em; d2.ldc = MSG_DIM; d2.M = NE_; d2.accum = (dir == 1);
    launch_gemm(d2, stream);
  }
  edge_plan_kernel<<<B_, 256, 0, stream>>>(plan, em, from_idx, to_idx, out);
}